// LoRA_47991964566282
// MI455X (gfx1250) — compile-verified
//
#include <hip/hip_runtime.h>

typedef __attribute__((ext_vector_type(2))) float v2f;
typedef __attribute__((ext_vector_type(8))) float v8f;

#define D_LATENT 2048
#define RANK 16
#define WAVES_PER_WG 8
#define THREADS (WAVES_PER_WG * 32)
#define TILES_PER_WG (WAVES_PER_WG / 2)          // wave pair per 16-token tile
#define TOK_PER_WG (TILES_PER_WG * 16)           // 64 tokens per workgroup

// out[t, e] = sum_r ( sum_d h[t,d] * Bmat[r,d] ) * Amat[e,r]
__global__ __launch_bounds__(THREADS)
void lora_fused_wmma(const float* __restrict__ h,
                     const float* __restrict__ Amat,   // [D_LATENT, RANK] row-major
                     const float* __restrict__ Bmat,   // [RANK, D_LATENT] row-major
                     float* __restrict__ out)          // [tokens, D_LATENT]
{
    // Amat staged in LDS, pre-permuted into exact WMMA B-fragment order:
    // block (c = e>>4, q = k>>2): 32 lanes x 2 floats, contiguous, conflict-free.
    __shared__ float ldsA[D_LATENT * RANK];                 // 128 KB
    __shared__ float ldsT[TILES_PER_WG * 2 * 256];          // 8 KB partial-t scratch

    const int tid   = threadIdx.x;
    const int wave  = tid >> 5;
    const int lane  = tid & 31;
    const int ln    = lane & 15;
    const int lh    = lane >> 4;          // 0 = lanes 0-15, 1 = lanes 16-31
    const int tile  = wave >> 1;          // which 16-token tile within WG
    const int dhalf = wave & 1;           // K-split half of d (and e-half in stage 2)

    // ---------- Async-stage Amat -> LDS in B-fragment order ----------
    // pair p covers Amat[e][2*kp .. 2*kp+1]; e = p>>3, kp = p&7  (RANK/2 = 8 pairs/row)
    // dest: block (e>>4, kp>>1), slot n = e&15 (+16 if kp odd)
    for (int p = tid; p < (D_LATENT * RANK) / 2; p += THREADS) {
        const int e  = p >> 3;
        const int kp = p & 7;
        const float* gaddr = Amat + e * RANK + 2 * kp;
        const unsigned lds_off =
            (unsigned)(uintptr_t)(ldsA +
                (((e >> 4) * 4 + (kp >> 1)) * 32 + (e & 15) + 16 * (kp & 1)) * 2);
        asm volatile("global_load_async_to_lds_b64 %0, %1, off"
                     :: "v"(lds_off), "v"(gaddr) : "memory");
    }

    // ---------- Stage 1 (K-split): partial t[m][r] over d in [dhalf*1024, +1024) ----
    // A-frag (16x4 f32): lane = M (token), VGPR0/1 = K {0,1} (lo) / {2,3} (hi)
    // B-frag (4x16 f32): lane = N (rank r), VGPR0/1 = K {0,1} (lo) / {2,3} (hi)
    const long tok0 = (long)blockIdx.x * TOK_PER_WG + (long)tile * 16;
    const int  dbase = dhalf * (D_LATENT / 2);
    const float* hrow = h    + (tok0 + ln) * D_LATENT + 2 * lh + dbase;
    const float* brow = Bmat + (long)ln   * D_LATENT + 2 * lh + dbase;

    v8f acc0 = {}, acc1 = {}, acc2 = {}, acc3 = {};
    for (int d0 = 0; d0 < D_LATENT / 2; d0 += 16) {
        v2f a, b;
        a = *(const v2f*)(hrow + d0);      b = *(const v2f*)(brow + d0);
        acc0 = __builtin_amdgcn_wmma_f32_16x16x4_f32(false, a, false, b, (short)0, acc0, false, false);
        a = *(const v2f*)(hrow + d0 + 4);  b = *(const v2f*)(brow + d0 + 4);
        acc1 = __builtin_amdgcn_wmma_f32_16x16x4_f32(false, a, false, b, (short)0, acc1, false, false);
        a = *(const v2f*)(hrow + d0 + 8);  b = *(const v2f*)(brow + d0 + 8);
        acc2 = __builtin_amdgcn_wmma_f32_16x16x4_f32(false, a, false, b, (short)0, acc2, false, false);
        a = *(const v2f*)(hrow + d0 + 12); b = *(const v2f*)(brow + d0 + 12);
        acc3 = __builtin_amdgcn_wmma_f32_16x16x4_f32(false, a, false, b, (short)0, acc3, false, false);
    }
    const v8f tacc = (acc0 + acc1) + (acc2 + acc3);

    // ---------- Spill partial t: C-fragment -> LDS [m][r] ----------
    // C layout: VGPR j, lanes 0-15: (M=j, N=lane); lanes 16-31: (M=j+8, N=lane-16)
    float* tbuf = ldsT + (tile * 2 + dhalf) * 256;
#pragma unroll
    for (int j = 0; j < 8; ++j) {
        const int m = j + 8 * lh;
        tbuf[m * 16 + ln] = tacc[j];
    }

    // close our async staging, then synchronize the whole workgroup
    asm volatile("s_wait_asynccnt 0" ::: "memory");
    __syncthreads();

    // ---------- Build A2-frags: sum the two K-split halves of t ----------
    // A2-frag chunk q: lane holds t[M=ln][4q + 2*lh + {0,1}]
    const float* t0 = ldsT + (tile * 2 + 0) * 256;
    const float* t1 = ldsT + (tile * 2 + 1) * 256;
    v2f a2[4];
#pragma unroll
    for (int q = 0; q < 4; ++q) {
        const int idx = ln * 16 + q * 4 + 2 * lh;
        a2[q] = *(const v2f*)(t0 + idx) + *(const v2f*)(t1 + idx);
    }

    // ---------- Stage 2: out[m][e] = sum_r t[m][r] * Amat[e][r] ----------
    // wave pair splits the 128 e-chunks: this wave owns [dhalf*64, dhalf*64+64)
    const int cbase = dhalf * (D_LATENT / 32);
#pragma unroll 2
    for (int ci = 0; ci < D_LATENT / 32; ++ci) {
        const int c = cbase + ci;
        v8f oacc = {};
#pragma unroll
        for (int q = 0; q < 4; ++q) {
            const v2f b2 = *(const v2f*)(ldsA + ((c * 4 + q) * 32 + lane) * 2);
            oacc = __builtin_amdgcn_wmma_f32_16x16x4_f32(
                false, a2[q], false, b2, (short)0, oacc, false, false);
        }
        // C layout store: VGPR j -> out[tok0 + j + 8*lh][c*16 + ln]
#pragma unroll
        for (int j = 0; j < 8; ++j) {
            out[(tok0 + j + 8 * lh) * D_LATENT + c * 16 + ln] = oacc[j];
        }
    }
}

extern "C" void kernel_launch(void* const* d_in, const int* in_sizes, int n_in,
                              void* d_out, int out_size, void* d_ws, size_t ws_size,
                              hipStream_t stream) {
    const float* h    = (const float*)d_in[0];
    const float* Amat = (const float*)d_in[1];   // [2048, 16]
    const float* Bmat = (const float*)d_in[2];   // [16, 2048]
    float* out = (float*)d_out;

    const int tokens = in_sizes[0] / D_LATENT;   // 4*4096 = 16384
    const int blocks = tokens / TOK_PER_WG;      // 256

    lora_fused_wmma<<<blocks, THREADS, 0, stream>>>(h, Amat, Bmat, out);
}